// GATLayer_57071525429598
// MI455X (gfx1250) — compile-verified
//
#include <hip/hip_runtime.h>
#include <hip/hip_bf16.h>
#include <math.h>

// Problem constants (match reference)
#define B_      8
#define N_      1024
#define CIN_    256
#define H_      4
#define COUT_   256
#define C_      64          // per-head channels
#define ALPHA_  0.2f
#define NEG_INF_ (-9.0e15f)

typedef float v2f __attribute__((ext_vector_type(2)));
typedef float v8f __attribute__((ext_vector_type(8)));

// -------------------------------------------------------------------------
// Kernel 1: h = X @ W^T + b   (M=B*N=8192, N=COUT=256, K=CIN=256)
// One wave per 16x64 output slab: 4 accumulators share one A fragment per
// K-step, V_WMMA_F32_16X16X4_F32 with K-step 4.
// A (16x4 f32) lane layout: lanes 0-15 hold {K=0,K=1}, lanes 16-31 {K=2,K=3}
// B (4x16 f32) mirrored:    v0 = rows {K=0 | K=2}, v1 = rows {K=1 | K=3}
// D (16x16 f32): VGPR r -> row r (lanes 0-15) / row r+8 (lanes 16-31)
// -------------------------------------------------------------------------
__global__ __launch_bounds__(256) void gat_gemm_h(const float* __restrict__ X,
                                                  const float* __restrict__ W,
                                                  const float* __restrict__ bias,
                                                  float* __restrict__ hbuf) {
  const int wave = blockIdx.x * (blockDim.x >> 5) + (threadIdx.x >> 5);
  const int lane = threadIdx.x & 31;
  const int half = lane >> 4;     // 0: K pair {0,1}, 1: K pair {2,3}
  const int l    = lane & 15;

  const int tileM = wave >> 2;    // 512 row tiles of 16
  const int cg    = wave & 3;     // 4 column groups of 64
  const int row0  = tileM * 16;
  const int col0  = cg * 64;

  const float* xrow = X + (size_t)(row0 + l) * CIN_;           // A row M=l
  const float* wr0  = W + (size_t)(col0 +  0 + l) * CIN_;      // B col n=l of chunk cc
  const float* wr1  = W + (size_t)(col0 + 16 + l) * CIN_;
  const float* wr2  = W + (size_t)(col0 + 32 + l) * CIN_;
  const float* wr3  = W + (size_t)(col0 + 48 + l) * CIN_;

  v8f acc[4] = {{}, {}, {}, {}};
  #pragma unroll 4
  for (int kk = 0; kk < CIN_; kk += 4) {
    const int ko = kk + 2 * half;
    v2f av  = *(const v2f*)(xrow + ko);
    v2f bv0 = *(const v2f*)(wr0 + ko);
    v2f bv1 = *(const v2f*)(wr1 + ko);
    v2f bv2 = *(const v2f*)(wr2 + ko);
    v2f bv3 = *(const v2f*)(wr3 + ko);
    acc[0] = __builtin_amdgcn_wmma_f32_16x16x4_f32(false, av, false, bv0,
                                                   (short)0, acc[0], false, false);
    acc[1] = __builtin_amdgcn_wmma_f32_16x16x4_f32(false, av, false, bv1,
                                                   (short)0, acc[1], false, false);
    acc[2] = __builtin_amdgcn_wmma_f32_16x16x4_f32(false, av, false, bv2,
                                                   (short)0, acc[2], false, false);
    acc[3] = __builtin_amdgcn_wmma_f32_16x16x4_f32(false, av, false, bv3,
                                                   (short)0, acc[3], false, false);
  }

  #pragma unroll
  for (int cc = 0; cc < 4; ++cc) {
    const float bval = bias[col0 + cc * 16 + l];
    #pragma unroll
    for (int r = 0; r < 8; ++r) {
      const int row = row0 + r + 8 * half;
      hbuf[(size_t)row * COUT_ + col0 + cc * 16 + l] = acc[cc][r] + bval;
    }
  }
}

// -------------------------------------------------------------------------
// Kernel 2: l_src[b,i,h] = h[b,i,h,:]·a_src[h,:],  l_dst likewise.
// One thread per (b,i,head).
// -------------------------------------------------------------------------
__global__ __launch_bounds__(256) void gat_lscore(const float* __restrict__ hbuf,
                                                  const float* __restrict__ a,
                                                  float* __restrict__ lsrc,
                                                  float* __restrict__ ldst) {
  const int t    = blockIdx.x * blockDim.x + threadIdx.x;  // 0 .. B*N*H-1
  const int head = t & (H_ - 1);
  const int bi   = t >> 2;                                  // b*N + i
  const float* hr = hbuf + (size_t)bi * COUT_ + head * C_;
  const float* as = a + head * (2 * C_);
  const float* ad = as + C_;
  float s = 0.f, d = 0.f;
  #pragma unroll 8
  for (int c = 0; c < C_; ++c) {
    const float v = hr[c];
    s = fmaf(v, as[c], s);
    d = fmaf(v, ad[c], d);
  }
  lsrc[t] = s;
  ldst[t] = d;
}

// -------------------------------------------------------------------------
// Kernel 3: per (b,i,head) masked-softmax row statistics:
//   m = max_j masked(lrelu(ls_i + ld_j)),  s = sum_j exp(logit - m)
// One wave per (b,i); wave32 shuffle reductions.
// -------------------------------------------------------------------------
__global__ __launch_bounds__(256) void gat_stats(const float* __restrict__ lsrc,
                                                 const float* __restrict__ ldst,
                                                 const int*   __restrict__ adj,
                                                 float* __restrict__ mbuf,
                                                 float* __restrict__ sbuf) {
  const int wave = blockIdx.x * (blockDim.x >> 5) + (threadIdx.x >> 5);
  const int lane = threadIdx.x & 31;
  const size_t bi = (size_t)wave;              // b*N + i
  const int b = wave / N_;

  float ls[H_];
  #pragma unroll
  for (int hd = 0; hd < H_; ++hd) ls[hd] = lsrc[bi * H_ + hd];

  const int*    arow = adj + bi * (size_t)N_;
  const float4* ldr  = (const float4*)ldst;    // H_==4 contiguous per node

  float mx[H_];
  #pragma unroll
  for (int hd = 0; hd < H_; ++hd) mx[hd] = NEG_INF_;

  // pass 1: row max
  for (int j = lane; j < N_; j += 32) {
    const int    ad = arow[j];
    const float4 ld = ldr[(size_t)b * N_ + j];
    const float  ldv[H_] = {ld.x, ld.y, ld.z, ld.w};
    #pragma unroll
    for (int hd = 0; hd < H_; ++hd) {
      float x = ls[hd] + ldv[hd];
      x = fmaxf(x, ALPHA_ * x);                // leaky relu (alpha<1)
      x = (ad == 1) ? x : NEG_INF_;
      mx[hd] = fmaxf(mx[hd], x);
    }
  }
  #pragma unroll
  for (int hd = 0; hd < H_; ++hd)
    #pragma unroll
    for (int off = 16; off > 0; off >>= 1)
      mx[hd] = fmaxf(mx[hd], __shfl_xor(mx[hd], off, 32));

  // pass 2: exp-sum
  float sm[H_] = {0.f, 0.f, 0.f, 0.f};
  for (int j = lane; j < N_; j += 32) {
    const int    ad = arow[j];
    const float4 ld = ldr[(size_t)b * N_ + j];
    const float  ldv[H_] = {ld.x, ld.y, ld.z, ld.w};
    #pragma unroll
    for (int hd = 0; hd < H_; ++hd) {
      float x = ls[hd] + ldv[hd];
      x = fmaxf(x, ALPHA_ * x);
      sm[hd] += (ad == 1) ? __expf(x - mx[hd]) : 0.f;
    }
  }
  #pragma unroll
  for (int hd = 0; hd < H_; ++hd)
    #pragma unroll
    for (int off = 16; off > 0; off >>= 1)
      sm[hd] += __shfl_xor(sm[hd], off, 32);

  if (lane == 0) {
    #pragma unroll
    for (int hd = 0; hd < H_; ++hd) {
      mbuf[bi * H_ + hd] = mx[hd];
      sbuf[bi * H_ + hd] = sm[hd];
    }
  }
}

// -------------------------------------------------------------------------
// Kernel 4: out[b,i,head,:] = sum_j P[b,i,j,head] * h[b,j,head,:]
// One wave per (b, head, 32-row i-slab): two 16-row D tiles share every B
// (h) fragment, halving h traffic per WMMA. l_dst for the whole batch row
// block (16 KB) is staged in LDS once per block and broadcast-read.
// Probabilities are materialized straight into the WMMA A operand:
//   P = exp(lrelu(ls_i + ld_j) - m_i) / s_i   (0 where adj != 1)
// -------------------------------------------------------------------------
__global__ __launch_bounds__(128) void gat_aggregate(const float* __restrict__ hbuf,
                                                     const float* __restrict__ lsrc,
                                                     const float* __restrict__ ldst,
                                                     const int*   __restrict__ adj,
                                                     const float* __restrict__ mbuf,
                                                     const float* __restrict__ sbuf,
                                                     float* __restrict__ out) {
  __shared__ float lds_ld[N_ * H_];           // l_dst[b, j, head], 16 KB

  const int b     = blockIdx.x >> 5;          // / (N/32)
  const int islab = blockIdx.x & 31;
  const int i0    = islab * 32;
  const int head  = threadIdx.x >> 5;
  const int lane  = threadIdx.x & 31;
  const int half  = lane >> 4;
  const int l     = lane & 15;

  // cooperative stage of l_dst[b,:,:] into LDS (float4 per node)
  {
    const float4* src = (const float4*)(ldst + (size_t)b * N_ * H_);
    float4*       dst = (float4*)lds_ld;
    for (int j = threadIdx.x; j < N_; j += 128) dst[j] = src[j];
  }
  __syncthreads();

  const size_t biA = (size_t)b * N_ + (i0 + l);        // rows of tile A
  const size_t biB = biA + 16;                          // rows of tile B
  const float  lsA = lsrc[biA * H_ + head];
  const float  lsB = lsrc[biB * H_ + head];
  const float  mA  = mbuf[biA * H_ + head];
  const float  mB  = mbuf[biB * H_ + head];
  const float  isA = 1.0f / sbuf[biA * H_ + head];
  const float  isB = 1.0f / sbuf[biB * H_ + head];
  const int*   arA = adj + biA * (size_t)N_;
  const int*   arB = adj + biB * (size_t)N_;

  v8f accA[4] = {{}, {}, {}, {}};
  v8f accB[4] = {{}, {}, {}, {}};

  for (int j0 = 0; j0 < N_; j0 += 4) {
    const int ja = j0 + 2 * half;               // this lane's K pair {ja, ja+1}

    // --- A operands: normalized attention probabilities for both i-tiles
    const int2  adA = *(const int2*)(arA + ja);
    const int2  adB = *(const int2*)(arB + ja);
    const float ld0 = lds_ld[ja * H_ + head];
    const float ld1 = lds_ld[(ja + 1) * H_ + head];

    float xA0 = lsA + ld0; xA0 = fmaxf(xA0, ALPHA_ * xA0);
    float xA1 = lsA + ld1; xA1 = fmaxf(xA1, ALPHA_ * xA1);
    float xB0 = lsB + ld0; xB0 = fmaxf(xB0, ALPHA_ * xB0);
    float xB1 = lsB + ld1; xB1 = fmaxf(xB1, ALPHA_ * xB1);

    v2f avA, avB;
    avA.x = (adA.x == 1) ? __expf(xA0 - mA) * isA : 0.0f;
    avA.y = (adA.y == 1) ? __expf(xA1 - mA) * isA : 0.0f;
    avB.x = (adB.x == 1) ? __expf(xB0 - mB) * isB : 0.0f;
    avB.y = (adB.y == 1) ? __expf(xB1 - mB) * isB : 0.0f;

    // --- shared B operand: h rows j0+2*half (v0) and j0+2*half+1 (v1)
    const float* hrow0 = hbuf + ((size_t)b * N_ + ja) * COUT_ + head * C_ + l;
    const float* hrow1 = hrow0 + COUT_;
    #pragma unroll
    for (int cc = 0; cc < 4; ++cc) {
      v2f bv;
      bv.x = hrow0[cc * 16];
      bv.y = hrow1[cc * 16];
      accA[cc] = __builtin_amdgcn_wmma_f32_16x16x4_f32(false, avA, false, bv,
                                                       (short)0, accA[cc],
                                                       false, false);
      accB[cc] = __builtin_amdgcn_wmma_f32_16x16x4_f32(false, avB, false, bv,
                                                       (short)0, accB[cc],
                                                       false, false);
    }
  }

  // --- store D tiles: row = i0(+16) + r + 8*half, col = head*64 + cc*16 + l
  #pragma unroll
  for (int cc = 0; cc < 4; ++cc) {
    #pragma unroll
    for (int r = 0; r < 8; ++r) {
      const int rowA = i0 + r + 8 * half;
      const int rowB = rowA + 16;
      out[((size_t)b * N_ + rowA) * COUT_ + head * C_ + cc * 16 + l] = accA[cc][r];
      out[((size_t)b * N_ + rowB) * COUT_ + head * C_ + cc * 16 + l] = accB[cc][r];
    }
  }
}

// -------------------------------------------------------------------------
extern "C" void kernel_launch(void* const* d_in, const int* in_sizes, int n_in,
                              void* d_out, int out_size, void* d_ws, size_t ws_size,
                              hipStream_t stream) {
  const float* node_feats = (const float*)d_in[0];   // (B,N,CIN) f32
  const int*   adj        = (const int*)  d_in[1];   // (B,N,N)   i32
  const float* W          = (const float*)d_in[2];   // (COUT,CIN)
  const float* bias       = (const float*)d_in[3];   // (COUT,)
  const float* a          = (const float*)d_in[4];   // (H, 2C)
  float*       out        = (float*)d_out;           // (B,N,COUT)

  // workspace carve-up (floats)
  float* hbuf = (float*)d_ws;                               // B*N*COUT
  float* lsrc = hbuf + (size_t)B_ * N_ * COUT_;             // B*N*H
  float* ldst = lsrc + (size_t)B_ * N_ * H_;
  float* mbuf = ldst + (size_t)B_ * N_ * H_;
  float* sbuf = mbuf + (size_t)B_ * N_ * H_;

  // 1) projection GEMM: 2048 slab-waves (16x64 each), 8 waves/block
  gat_gemm_h<<<(B_ * N_ / 16) * (COUT_ / 64) / 8, 256, 0, stream>>>(
      node_feats, W, bias, hbuf);

  // 2) attention logit halves: B*N*H threads
  gat_lscore<<<(B_ * N_ * H_) / 256, 256, 0, stream>>>(hbuf, a, lsrc, ldst);

  // 3) softmax row stats: one wave per (b,i)
  gat_stats<<<(B_ * N_) / 8, 256, 0, stream>>>(lsrc, ldst, adj, mbuf, sbuf);

  // 4) WMMA aggregation: one block per (b, 32-row i-slab), 4 head-waves
  gat_aggregate<<<B_ * (N_ / 32), 128, 0, stream>>>(
      hbuf, lsrc, ldst, adj, mbuf, sbuf, out);
}